// SupConLossMemory_20856361190091
// MI455X (gfx1250) — compile-verified
//
#include <hip/hip_runtime.h>
#include <hip/hip_bf16.h>

typedef __attribute__((ext_vector_type(2))) float v2f;
typedef __attribute__((ext_vector_type(8))) float v8f;

#define N_Q 2048
#define M_B 65536
#define F_DIM 128
#define N_CLS 13
#define KNN 10
#define INV_TEMP (1.0f / 0.07f)
#define NEG_INF (-3.402823466e38f)
#define POS_INF (3.402823466e38f)

// ---------------------------------------------------------------------------
// 16x16 fp32 dot-product tile: D += A(16xK=128) * B(K=128 x 16), via 32x
// V_WMMA_F32_16X16X4_F32. A rows come from LDS (16x128), B columns are rows
// of a global row-major matrix starting at Bbase (col n == row n of Bbase).
// Per ISA 7.12.2 (wave32): A 16x4 f32 -> 2 VGPRs: lanes 0-15 hold K=0,1,
// lanes 16-31 hold K=2,3 (M = lane&15). B 4x16 mirrored. C/D: VGPR r ->
// M = r + 8*(lane>=16), N = lane&15.
// ---------------------------------------------------------------------------
__device__ __forceinline__ v8f wmma_dot_tile(const float* __restrict__ Asm,
                                             const float* __restrict__ Bbase,
                                             int lane) {
  const int h = lane >> 4;       // half select -> K pair offset
  const int m = lane & 15;       // A row / B column
  const float* brow = Bbase + (size_t)m * F_DIM;
  v8f c = {};
#pragma unroll
  for (int k0 = 0; k0 < F_DIM; k0 += 4) {
    v2f a, b;
    a.x = Asm[m * F_DIM + k0 + 2 * h];
    a.y = Asm[m * F_DIM + k0 + 2 * h + 1];
    b.x = brow[k0 + 2 * h];
    b.y = brow[k0 + 2 * h + 1];
    c = __builtin_amdgcn_wmma_f32_16x16x4_f32(false, a, false, b, (short)0, c,
                                              false, false);
  }
  return c;
}

// ---------------------------------------------------------------------------
// Row-norm precompute: bN[i] = ||bank_i||^2, qN[i] = ||feat_i||^2
// ---------------------------------------------------------------------------
__global__ __launch_bounds__(256) void norms_kernel(
    const float* __restrict__ feats, const float* __restrict__ bank,
    float* __restrict__ qN, float* __restrict__ bN) {
  int tid = blockIdx.x * blockDim.x + threadIdx.x;
  const float* src;
  float* dst;
  if (tid < M_B) {
    src = bank + (size_t)tid * F_DIM;
    dst = bN + tid;
  } else if (tid < M_B + N_Q) {
    src = feats + (size_t)(tid - M_B) * F_DIM;
    dst = qN + (tid - M_B);
  } else {
    return;
  }
  const float4* p4 = (const float4*)src;
  float s = 0.f;
#pragma unroll 4
  for (int i = 0; i < F_DIM / 4; ++i) {
    float4 v = p4[i];
    s += v.x * v.x + v.y * v.y + v.z * v.z + v.w * v.w;
  }
  *dst = s;
}

// ---------------------------------------------------------------------------
// SupCon loss: one block = 16 anchor rows. Two passes over all 2048 columns
// (pass 1: row max; pass 2: Z = sum off-diag exp, P = masked logit sum,
// C = positive count). contrib_i = -(P - C*m - C*log Z)/max(C,1).
// ---------------------------------------------------------------------------
__global__ __launch_bounds__(256) void supcon_kernel(
    const float* __restrict__ feats, const int* __restrict__ labels,
    float* __restrict__ out) {
  __shared__ float As[16 * F_DIM];
  __shared__ int Ls[N_Q];
  __shared__ float wred[8][16];
  __shared__ float wZ[8][16], wP[8][16], wC[8][16];
  __shared__ float rowMax[16];

  const int tid = threadIdx.x;
  const int wave = tid >> 5;
  const int lane = tid & 31;
  const int rowbase = blockIdx.x * 16;
  const int h = lane >> 4;
  const int n = lane & 15;

  for (int i = tid; i < 16 * F_DIM; i += 256)
    As[i] = feats[(size_t)rowbase * F_DIM + i];
  for (int i = tid; i < N_Q; i += 256) Ls[i] = labels[i];
  __syncthreads();

  // ---- pass 1: per-row max (includes diagonal, as in reference) ----
  float rmax[8];
#pragma unroll
  for (int r = 0; r < 8; ++r) rmax[r] = NEG_INF;

  for (int t = 0; t < 16; ++t) {
    const int colbase = (wave * 16 + t) * 16;
    v8f c = wmma_dot_tile(As, feats + (size_t)colbase * F_DIM, lane);
#pragma unroll
    for (int r = 0; r < 8; ++r) {
      float v = c[r] * INV_TEMP;
#pragma unroll
      for (int mk = 1; mk < 16; mk <<= 1) v = fmaxf(v, __shfl_xor(v, mk, 32));
      rmax[r] = fmaxf(rmax[r], v);
    }
  }
  if (n == 0) {
#pragma unroll
    for (int r = 0; r < 8; ++r) wred[wave][h * 8 + r] = rmax[r];
  }
  __syncthreads();
  if (tid < 16) {
    float mm = NEG_INF;
    for (int w = 0; w < 8; ++w) mm = fmaxf(mm, wred[w][tid]);
    rowMax[tid] = mm;
  }
  __syncthreads();

  // ---- pass 2: Z, P, C ----
  float rZ[8], rP[8], rC[8];
#pragma unroll
  for (int r = 0; r < 8; ++r) { rZ[r] = 0.f; rP[r] = 0.f; rC[r] = 0.f; }

  for (int t = 0; t < 16; ++t) {
    const int colbase = (wave * 16 + t) * 16;
    v8f c = wmma_dot_tile(As, feats + (size_t)colbase * F_DIM, lane);
#pragma unroll
    for (int r = 0; r < 8; ++r) {
      const int lr = r + 8 * h;
      const int grow = rowbase + lr;
      const int gcol = colbase + n;
      float v = c[r] * INV_TEMP;
      const bool diag = (grow == gcol);
      float e = diag ? 0.f : __expf(v - rowMax[lr]);
      float mt = (!diag && (Ls[grow] == Ls[gcol])) ? 1.f : 0.f;
      float p = mt * v;
#pragma unroll
      for (int mk = 1; mk < 16; mk <<= 1) {
        e += __shfl_xor(e, mk, 32);
        p += __shfl_xor(p, mk, 32);
        mt += __shfl_xor(mt, mk, 32);
      }
      rZ[r] += e;
      rP[r] += p;
      rC[r] += mt;
    }
  }
  if (n == 0) {
#pragma unroll
    for (int r = 0; r < 8; ++r) {
      wZ[wave][h * 8 + r] = rZ[r];
      wP[wave][h * 8 + r] = rP[r];
      wC[wave][h * 8 + r] = rC[r];
    }
  }
  __syncthreads();
  if (tid < 16) {
    float Z = 0.f, P = 0.f, C = 0.f;
    for (int w = 0; w < 8; ++w) {
      Z += wZ[w][tid];
      P += wP[w][tid];
      C += wC[w][tid];
    }
    const float m = rowMax[tid];
    const float denom = (C > 0.f) ? C : 1.f;
    const float contrib = -(P - C * m - C * __logf(Z)) / denom;
    atomicAdd(out, contrib);
  }
}

// ---------------------------------------------------------------------------
// kNN distances + per-chunk top-10. Block = 16 query rows x 4096 bank cols.
// 8 sub-chunks of 512 cols: WMMA dot tiles -> dist in LDS -> 16 threads/row
// keep register-resident sorted top-10 -> 16-way merge -> ws partials.
// ---------------------------------------------------------------------------
__global__ __launch_bounds__(256) void knn_dist_kernel(
    const float* __restrict__ feats, const float* __restrict__ bank,
    const float* __restrict__ qNorm, const float* __restrict__ bNorm,
    float* __restrict__ partD, int* __restrict__ partI) {
  __shared__ float Qs[16 * F_DIM];
  __shared__ float qn[16];
  __shared__ float Ds[16][512];
  __shared__ float candD[16][16][KNN];
  __shared__ int candI[16][16][KNN];

  const int tid = threadIdx.x;
  const int wave = tid >> 5;
  const int lane = tid & 31;
  const int rowbase = blockIdx.x * 16;
  const int chunkBase = blockIdx.y * 4096;
  const int h = lane >> 4;
  const int nn = lane & 15;
  const int selRow = tid >> 4;
  const int selLane = tid & 15;

  for (int i = tid; i < 16 * F_DIM; i += 256)
    Qs[i] = feats[(size_t)rowbase * F_DIM + i];
  if (tid < 16) qn[tid] = qNorm[rowbase + tid];
  __syncthreads();

  float td[KNN];
  int ti[KNN];
#pragma unroll
  for (int k = 0; k < KNN; ++k) { td[k] = POS_INF; ti[k] = 0x7fffffff; }

  for (int sub = 0; sub < 8; ++sub) {
    const int colbase0 = chunkBase + sub * 512;
#pragma unroll
    for (int tt = 0; tt < 4; ++tt) {
      const int colbase = colbase0 + (wave * 4 + tt) * 16;
      v8f c = wmma_dot_tile(Qs, bank + (size_t)colbase * F_DIM, lane);
#pragma unroll
      for (int r = 0; r < 8; ++r) {
        const int lr = r + 8 * h;
        const int col = colbase + nn;
        float sq = qn[lr] + bNorm[col] - 2.0f * c[r];
        Ds[lr][col - colbase0] = __fsqrt_rn(fmaxf(sq, 0.f));
      }
    }
    __syncthreads();
    // selection: thread (selRow, selLane) scans 32 columns
    for (int j = 0; j < 32; ++j) {
      const int lc = selLane * 32 + j;
      const float d = Ds[selRow][lc];
      const int gidx = colbase0 + lc;
      bool ins = (d < td[KNN - 1]) || (d == td[KNN - 1] && gidx < ti[KNN - 1]);
      if (ins) {
        int p = KNN - 1;
#pragma unroll
        for (int k = KNN - 2; k >= 0; --k) {
          bool mv = (d < td[k]) || (d == td[k] && gidx < ti[k]);
          if (mv) {
            td[k + 1] = td[k];
            ti[k + 1] = ti[k];
            p = k;
          }
        }
#pragma unroll
        for (int k = 0; k < KNN; ++k)
          if (k == p) {
            td[k] = d;
            ti[k] = gidx;
          }
      }
    }
    __syncthreads();
  }

#pragma unroll
  for (int k = 0; k < KNN; ++k) {
    candD[selRow][selLane][k] = td[k];
    candI[selRow][selLane][k] = ti[k];
  }
  __syncthreads();

  if (tid < 16) {  // 16-way merge of sorted candidate lists for row `tid`
    int pp[16];
    for (int s = 0; s < 16; ++s) pp[s] = 0;
    const int grow = rowbase + tid;
    for (int k = 0; k < KNN; ++k) {
      float bd = POS_INF;
      int bi = 0x7fffffff, bs = 0;
      for (int s = 0; s < 16; ++s) {
        if (pp[s] < KNN) {
          float d = candD[tid][s][pp[s]];
          int idx = candI[tid][s][pp[s]];
          if (d < bd || (d == bd && idx < bi)) {
            bd = d;
            bi = idx;
            bs = s;
          }
        }
      }
      pp[bs]++;
      partD[((size_t)grow * 16 + blockIdx.y) * KNN + k] = bd;
      partI[((size_t)grow * 16 + blockIdx.y) * KNN + k] = bi;
    }
  }
}

// ---------------------------------------------------------------------------
// Merge 16 per-chunk sorted top-10 lists -> final top-10 -> majority vote.
// ---------------------------------------------------------------------------
__global__ __launch_bounds__(256) void knn_vote_kernel(
    const float* __restrict__ partD, const int* __restrict__ partI,
    const int* __restrict__ bankLabels, const int* __restrict__ qLabels,
    float* __restrict__ out) {
  const int row = blockIdx.x * blockDim.x + threadIdx.x;
  if (row >= N_Q) return;

  int pp[16];
  for (int s = 0; s < 16; ++s) pp[s] = 0;
  int cnt[N_CLS];
  for (int c = 0; c < N_CLS; ++c) cnt[c] = 0;

  for (int k = 0; k < KNN; ++k) {
    float bd = POS_INF;
    int bi = 0x7fffffff, bs = 0;
    for (int s = 0; s < 16; ++s) {
      if (pp[s] < KNN) {
        float d = partD[((size_t)row * 16 + s) * KNN + pp[s]];
        int idx = partI[((size_t)row * 16 + s) * KNN + pp[s]];
        if (d < bd || (d == bd && idx < bi)) {
          bd = d;
          bi = idx;
          bs = s;
        }
      }
    }
    pp[bs]++;
    int lbl = bankLabels[bi];
    if (lbl >= 0 && lbl < N_CLS) cnt[lbl]++;
  }

  int pred = 0, bc = -1;
  for (int c = 0; c < N_CLS; ++c)
    if (cnt[c] > bc) {
      bc = cnt[c];
      pred = c;
    }
  if (pred == qLabels[row]) atomicAdd(out, 100.0f / (float)N_Q);
}

// ---------------------------------------------------------------------------
extern "C" void kernel_launch(void* const* d_in, const int* in_sizes, int n_in,
                              void* d_out, int out_size, void* d_ws,
                              size_t ws_size, hipStream_t stream) {
  (void)in_sizes;
  (void)n_in;
  (void)ws_size;
  const float* feats = (const float*)d_in[0];      // [1,2048,128] f32
  const float* bank = (const float*)d_in[1];       // [1,65536,128] f32
  const int* qLabels = (const int*)d_in[2];        // [1,2048]
  const int* bLabels = (const int*)d_in[3];        // [1,65536]
  float* out = (float*)d_out;

  // workspace layout (all 4-byte elements, ~2.9 MB total)
  float* bN = (float*)d_ws;                        // 65536
  float* qN = bN + M_B;                            // 2048
  float* partD = qN + N_Q;                         // 2048*16*10
  int* partI = (int*)(partD + (size_t)N_Q * 16 * KNN);

  hipMemsetAsync(d_out, 0, sizeof(float) * (size_t)out_size, stream);
  norms_kernel<<<(M_B + N_Q) / 256, 256, 0, stream>>>(feats, bank, qN, bN);
  supcon_kernel<<<N_Q / 16, 256, 0, stream>>>(feats, qLabels, out);
  knn_dist_kernel<<<dim3(N_Q / 16, 16), 256, 0, stream>>>(feats, bank, qN, bN,
                                                          partD, partI);
  knn_vote_kernel<<<N_Q / 256, 256, 0, stream>>>(partD, partI, bLabels,
                                                 qLabels, out);
}